// Gate_28303834480969
// MI455X (gfx1250) — compile-verified
//
#include <hip/hip_runtime.h>
#include <hip/hip_bf16.h>

typedef float v2f __attribute__((ext_vector_type(2)));
typedef float v8f __attribute__((ext_vector_type(8)));

#define TOKENS        16384
#define DMODEL        2048
#define NEXPERT       64
#define WAVES_PER_WG  4
#define WG_THREADS    (WAVES_PER_WG * 32)
#define TOK_PER_WAVE  16
#define TOK_PER_WG    (WAVES_PER_WG * TOK_PER_WAVE)
#define KC            64      /* K-chunk staged in LDS                 */
#define WSTRIDE       68      /* padded LDS row stride for W chunk     */
#define SSTRIDE       65      /* padded LDS row stride for logit spill */

// ---- CDNA5 async memory->LDS path (inline asm: portable across toolchains) ----
__device__ __forceinline__ unsigned lds_off(const void* p) {
    // generic -> LDS(AS3) pointer, then ptrtoint => byte offset in LDS space
    return (unsigned)(unsigned long long)
           (__attribute__((address_space(3))) const void*)p;
}

__device__ __forceinline__ void async_load_b128(unsigned lds_byte, const float* g) {
    asm volatile("global_load_async_to_lds_b128 %0, %1, off"
                 :: "v"(lds_byte), "v"((unsigned long long)g)
                 : "memory");
}

__device__ __forceinline__ void wait_async0() {
    asm volatile("s_wait_asynccnt 0x0" ::: "memory");
}

__global__ __launch_bounds__(WG_THREADS)
void moe_gate_kernel(const float* __restrict__ x,
                     const float* __restrict__ w,
                     float* __restrict__ outv,
                     int*   __restrict__ outi)
{
    __shared__ float wlds[2][NEXPERT * WSTRIDE];           // 2 x 17,408 B
    __shared__ float slog[WAVES_PER_WG * 16 * SSTRIDE];    // 16,640 B

    const int tid  = threadIdx.x;
    const int lane = tid & 31;
    const int wave = tid >> 5;
    const int m    = lane & 15;   // row within 16 (A) / col within 16 (B)
    const int kh   = lane >> 4;   // K-half select for A/B fragments

    const long tok0 = (long)(blockIdx.x * WAVES_PER_WG + wave) * TOK_PER_WAVE;
    const float* xrow = x + (tok0 + m) * (long)DMODEL + kh * 2;

    // cooperative weight staging: 128 threads x 8 b128 = 64 experts x 64 K f32
    const int    se   = tid >> 1;           // expert row 0..63
    const int    sc   = (tid & 1) * 32;     // col base 0 or 32
    const float* wsrc = w + (long)se * DMODEL + sc;
    const unsigned ldsA = lds_off(&wlds[0][se * WSTRIDE + sc]);
    const unsigned ldsB = lds_off(&wlds[1][se * WSTRIDE + sc]);

    v8f acc[4] = {};   // 4 N-tiles of 16 experts, 16x16 f32 accum each

    // prologue: kick off chunk 0 into buffer 0
    #pragma unroll
    for (int j = 0; j < 8; ++j)
        async_load_b128(ldsA + 16 * j, wsrc + 4 * j);

    int buf = 0;
    for (int k0 = 0; k0 < DMODEL; k0 += KC) {
        wait_async0();        // this wave's async loads for current buffer done
        __syncthreads();      // all waves' loads done; other buffer now free

        if (k0 + KC < DMODEL) {
            // overlap: fetch next chunk into the other buffer while we compute
            const unsigned l = buf ? ldsA : ldsB;
            const float*   g = wsrc + k0 + KC;
            #pragma unroll
            for (int j = 0; j < 8; ++j)
                async_load_b128(l + 16 * j, g + 4 * j);
            __builtin_prefetch(xrow + k0 + KC, 0, 0);   // warm next x K-block
        }

        const float* wb = &wlds[buf][0];
        #pragma unroll 4
        for (int kk = 0; kk < KC; kk += 4) {
            v2f a = *(const v2f*)(xrow + k0 + kk);      // A: x[m, k+2kh+{0,1}]
            #pragma unroll
            for (int nt = 0; nt < 4; ++nt) {
                const v2f b = *(const v2f*)&wb[(nt * 16 + m) * WSTRIDE + kk + kh * 2];
                acc[nt] = __builtin_amdgcn_wmma_f32_16x16x4_f32(
                    /*neg_a=*/false, a, /*neg_b=*/false, b,
                    /*c_mod=*/(short)0, acc[nt],
                    /*reuse_a=*/false, /*reuse_b=*/false);
            }
        }
        buf ^= 1;
    }

    // ---- spill logits to LDS: row M = v + 8*kh, expert = nt*16 + m ----
    #pragma unroll
    for (int nt = 0; nt < 4; ++nt) {
        #pragma unroll
        for (int v = 0; v < 8; ++v) {
            slog[(wave * 16 + v + 8 * kh) * SSTRIDE + nt * 16 + m] = acc[nt][v];
        }
    }
    __syncthreads();

    // ---- per-token top-2 + renormalized softmax weights ----
    if (lane < 16) {
        const float* s = &slog[(wave * 16 + lane) * SSTRIDE];
        float l1 = -__builtin_inff(), l2 = -__builtin_inff();
        int   i1 = 0, i2 = 0;
        #pragma unroll 8
        for (int e = 0; e < NEXPERT; ++e) {
            float vv = s[e];
            if (vv > l1)      { l2 = l1; i2 = i1; l1 = vv; i1 = e; }
            else if (vv > l2) { l2 = vv; i2 = e; }
        }
        // softmax(top2)/sum(top2) == [1/(1+r), r/(1+r)], r = exp(l2-l1)
        float r   = __expf(l2 - l1);
        float inv = 1.0f / (1.0f + r);
        long  t   = tok0 + lane;
        outv[2 * t]     = inv;
        outv[2 * t + 1] = r * inv;
        outi[2 * t]     = i1;
        outi[2 * t + 1] = i2;
    }
}

extern "C" void kernel_launch(void* const* d_in, const int* in_sizes, int n_in,
                              void* d_out, int out_size, void* d_ws, size_t ws_size,
                              hipStream_t stream) {
    (void)in_sizes; (void)n_in; (void)out_size; (void)d_ws; (void)ws_size;
    const float* x = (const float*)d_in[0];
    const float* w = (const float*)d_in[1];
    float* outv = (float*)d_out;                 // [16384, 2] f32 values
    int*   outi = (int*)d_out + 2 * TOKENS;      // [16384, 2] i32 indices
    dim3 grid(TOKENS / TOK_PER_WG);              // 256 workgroups
    dim3 block(WG_THREADS);                      // 128 threads = 4 waves
    moe_gate_kernel<<<grid, block, 0, stream>>>(x, w, outv, outi);
}